// ScaledDotProductAttention_56573309224314
// MI455X (gfx1250) — compile-verified
//
#include <hip/hip_runtime.h>
#include <math.h>

typedef float v2f __attribute__((ext_vector_type(2)));
typedef float v8f __attribute__((ext_vector_type(8)));

namespace {
constexpr int kB  = 32;
constexpr int kSQ = 2048;
constexpr int kSK = 2048;
constexpr int kD  = 64;
constexpr int kKT = kSK / 16;   // 128 key tiles per row strip
constexpr int kQS = kSQ / 16;   // 128 query strips per batch
constexpr float kScale = 0.125f; // 1/sqrt(64)
}

// One wave (32 lanes) owns a 16-query-row strip of one batch.
// Pass 1: S = Q K^T via v_wmma_f32_16x16x4_f32, online row max/sum.
// Pass 2: recompute S, write P = softmax(S) (non-temporal), and accumulate
//         O = P V via WMMA, transposing P C-layout -> A-layout through LDS.
__global__ __launch_bounds__(128) void attn_fused(
    const float* __restrict__ Qp, const float* __restrict__ Kp,
    const float* __restrict__ Vp, const unsigned char* __restrict__ Mp,
    float* __restrict__ Pp, float* __restrict__ Op)
{
  __shared__ float ptile[4][16 * 16];   // 1 KB per wave
  const int lane = threadIdx.x & 31;
  const int wv   = threadIdx.x >> 5;
  const int wid  = blockIdx.x * 4 + wv;
  const int b    = wid / kQS;
  const int q0   = (wid - b * kQS) * 16;
  const int hi   = lane >> 4;   // which 16-lane half
  const int lm   = lane & 15;

  // ---- Q A-fragments (16 chunks of K=4), resident all kernel.
  // A layout: lane L holds M = L%16; VGPR j holds K = 4c + 2*(L/16) + j.
  v2f aq[16];
  {
    const float* qr = Qp + ((size_t)b * kSQ + q0 + lm) * kD + 2 * hi;
    #pragma unroll
    for (int c = 0; c < 16; ++c) aq[c] = *(const v2f*)(qr + 4 * c);
  }

  const float* krow = Kp + ((size_t)b * kSK + lm) * kD + 2 * hi;
  const unsigned char* mrow0 =
      Mp + ((size_t)b * kSQ + q0 + 8 * hi) * (size_t)kSK + lm;

  float rm[8], rs[8];
  #pragma unroll
  for (int v = 0; v < 8; ++v) { rm[v] = -__builtin_inff(); rs[v] = 0.f; }

  // ---------------- pass 1: online softmax statistics ----------------
  for (int kt = 0; kt < kKT; ++kt) {
    v8f c = {};
    const float* kr = krow + (size_t)kt * 16 * kD;
    #pragma unroll
    for (int cc = 0; cc < 16; ++cc) {
      v2f bk = *(const v2f*)(kr + 4 * cc);  // B: n=lane%16 keys, k=2*hi+j dims
      c = __builtin_amdgcn_wmma_f32_16x16x4_f32(false, aq[cc], false, bk,
                                                (short)0, c, false, false);
    }
    const unsigned char* mp = mrow0 + (size_t)kt * 16;
    #pragma unroll
    for (int v = 0; v < 8; ++v) {      // C layout: row M = v + 8*hi, N = lm
      float sv = c[v] * kScale;
      sv = mp[(size_t)v * kSK] ? sv : -1e9f;
      float mo = rm[v];
      float mn = fmaxf(mo, sv);
      rm[v] = mn;
      rs[v] = rs[v] * __expf(mo - mn) + __expf(sv - mn);
    }
  }

  // Combine partial (max,sum) across the 16 lanes sharing each matrix row.
  #pragma unroll
  for (int xm = 1; xm <= 8; xm <<= 1) {
    #pragma unroll
    for (int v = 0; v < 8; ++v) {
      float om = __shfl_xor(rm[v], xm, 32);
      float os = __shfl_xor(rs[v], xm, 32);
      float mn = fmaxf(rm[v], om);
      rs[v] = rs[v] * __expf(rm[v] - mn) + os * __expf(om - mn);
      rm[v] = mn;
    }
  }
  float rinv[8];
  #pragma unroll
  for (int v = 0; v < 8; ++v) rinv[v] = 1.0f / rs[v];

  // ---------------- pass 2: P out (streaming) + O = P V ----------------
  v8f acc[4] = {};
  float* lp = ptile[wv];
  for (int kt = 0; kt < kKT; ++kt) {
    v8f c = {};
    const float* kr = krow + (size_t)kt * 16 * kD;
    #pragma unroll
    for (int cc = 0; cc < 16; ++cc) {
      v2f bk = *(const v2f*)(kr + 4 * cc);
      c = __builtin_amdgcn_wmma_f32_16x16x4_f32(false, aq[cc], false, bk,
                                                (short)0, c, false, false);
    }
    const unsigned char* mp = mrow0 + (size_t)kt * 16;
    float* pout = Pp + ((size_t)b * kSQ + q0 + 8 * hi) * (size_t)kSK
                     + (size_t)kt * 16 + lm;
    #pragma unroll
    for (int v = 0; v < 8; ++v) {
      float sv = c[v] * kScale;
      sv = mp[(size_t)v * kSK] ? sv : -1e9f;
      float p = __expf(sv - rm[v]) * rinv[v];
      __builtin_nontemporal_store(p, pout + (size_t)v * kSK); // write-once data
      lp[(v + 8 * hi) * 16 + lm] = p;   // stash tile row-major [m][k] in LDS
    }
    // O += P_tile(16x16) * V_tile(16x64): K split into 4 chunks of 4.
    // DS ops from one wave are processed in order -> loads see the stores.
    const float* vb0 = Vp + ((size_t)b * kSK + (size_t)kt * 16 + 2 * hi) * kD + lm;
    #pragma unroll
    for (int cc = 0; cc < 4; ++cc) {
      v2f ap = *(const v2f*)&lp[lm * 16 + cc * 4 + 2 * hi]; // A layout of P
      const float* vb = vb0 + (size_t)cc * 4 * kD;
      #pragma unroll
      for (int nt = 0; nt < 4; ++nt) {
        v2f bv = { vb[nt * 16], vb[nt * 16 + kD] };  // k and k+1 rows of V
        acc[nt] = __builtin_amdgcn_wmma_f32_16x16x4_f32(false, ap, false, bv,
                                                        (short)0, acc[nt],
                                                        false, false);
      }
    }
  }

  // ---- store O (C layout: row q0 + v + 8*hi, col nt*16 + lm)
  #pragma unroll
  for (int nt = 0; nt < 4; ++nt) {
    #pragma unroll
    for (int v = 0; v < 8; ++v) {
      Op[((size_t)b * kSQ + q0 + v + 8 * hi) * kD + nt * 16 + lm] = acc[nt][v];
    }
  }
}

extern "C" void kernel_launch(void* const* d_in, const int* in_sizes, int n_in,
                              void* d_out, int out_size, void* d_ws, size_t ws_size,
                              hipStream_t stream) {
  (void)in_sizes; (void)n_in; (void)out_size; (void)d_ws; (void)ws_size;
  const float* Q = (const float*)d_in[0];
  const float* K = (const float*)d_in[1];
  const float* V = (const float*)d_in[2];
  const unsigned char* M = (const unsigned char*)d_in[3];  // jax bool = 1 byte
  float* P = (float*)d_out;                                // [B,SQ,SK] softmax
  float* O = P + (size_t)kB * kSQ * kSK;                   // [B,SQ,D]
  dim3 grid((kB * kQS) / 4);   // 4096 strips, 4 waves per block
  attn_fused<<<grid, 128, 0, stream>>>(Q, K, V, M, P, O);
}